// Classification_net_36292473651977
// MI455X (gfx1250) — compile-verified
//
#include <hip/hip_runtime.h>
#include <math.h>

typedef float v2f __attribute__((ext_vector_type(2)));
typedef float v8f __attribute__((ext_vector_type(8)));

#define B_DIM   64
#define C_OUT   43
#define C_PAD   48
#define K_DIM   128
#define S_DIM   2500      // 50*50
#define M_TOT   160000    // 64*2500
#define N_TILES 10000     // M_TOT/16

// ---------------------------------------------------------------------------
// Kernel 1: zero the union column mask
// ---------------------------------------------------------------------------
__global__ void init_mask_kernel(float* colmask) {
    int i = blockIdx.x * blockDim.x + threadIdx.x;
    if (i < S_DIM) colmask[i] = 0.0f;
}

// ---------------------------------------------------------------------------
// Kernel 2: fused 1x1conv -> relu -> 1x1conv -> relu, via V_WMMA_F32_16X16X4_F32
// One wave (32 lanes) per 16-position M-tile. 8 waves / block, 1250 blocks.
// ---------------------------------------------------------------------------
__global__ void __launch_bounds__(256) mlp_wmma_kernel(
    const float* __restrict__ x,   // (64,3,2500)
    const float* __restrict__ w1,  // (128,3)
    const float* __restrict__ b1,  // (128,)
    const float* __restrict__ w2,  // (43,128)
    const float* __restrict__ b2,  // (43,)
    float* __restrict__ Y)         // (64,43,2500)
{
    __shared__ float w2s[C_PAD * K_DIM]; // padded to 48 rows, zeros beyond 43
    __shared__ float w1s[K_DIM * 3];
    __shared__ float b1s[K_DIM];
    __shared__ float b2s[C_PAD];

    // ---- stage weights into LDS (uniform, no divergence hazards for WMMA) ----
    for (int i = threadIdx.x; i < C_PAD * K_DIM; i += 256) {
        int n = i / K_DIM, k = i % K_DIM;
        w2s[i] = (n < C_OUT) ? w2[n * K_DIM + k] : 0.0f;
    }
    for (int i = threadIdx.x; i < K_DIM * 3; i += 256) w1s[i] = w1[i];
    for (int i = threadIdx.x; i < K_DIM; i += 256)     b1s[i] = b1[i];
    for (int i = threadIdx.x; i < C_PAD; i += 256)     b2s[i] = (i < C_OUT) ? b2[i] : 0.0f;
    __syncthreads();

    const int wave = threadIdx.x >> 5;
    const int lane = threadIdx.x & 31;
    const int tile = blockIdx.x * 8 + wave;     // 0..9999, always valid

    const int m   = lane & 15;                  // row of A this lane feeds
    const int off = (lane < 16) ? 0 : 2;        // K sub-offset per A/B layout

    // ---- first conv (K=3) computed directly in WMMA A-layout registers ----
    const int p = tile * 16 + m;
    const int b = p / S_DIM;
    const int s = p % S_DIM;
    const float x0 = x[(b * 3 + 0) * S_DIM + s];
    const float x1 = x[(b * 3 + 1) * S_DIM + s];
    const float x2 = x[(b * 3 + 2) * S_DIM + s];

    float hreg[64];
#pragma unroll
    for (int c = 0; c < 32; ++c) {
#pragma unroll
        for (int j = 0; j < 2; ++j) {
            const int k = 4 * c + off + j;
            float v = fmaf(x0, w1s[k * 3 + 0],
                      fmaf(x1, w1s[k * 3 + 1],
                      fmaf(x2, w1s[k * 3 + 2], b1s[k])));
            hreg[2 * c + j] = fmaxf(v, 0.0f);
        }
    }

    // ---- second conv: 32 chained K=4 WMMA steps, 3 N-tiles (48 cols) ----
    const int n0 = lane & 15;
    v8f acc0 = {}; v8f acc1 = {}; v8f acc2 = {};
#pragma unroll
    for (int c = 0; c < 32; ++c) {
        const int kk = 4 * c + off;
        v2f a; a.x = hreg[2 * c]; a.y = hreg[2 * c + 1];
        // contiguous, 8-byte aligned -> ds_load_b64
        v2f bt0 = *(const v2f*)&w2s[(0  + n0) * K_DIM + kk];
        v2f bt1 = *(const v2f*)&w2s[(16 + n0) * K_DIM + kk];
        v2f bt2 = *(const v2f*)&w2s[(32 + n0) * K_DIM + kk];
        acc0 = __builtin_amdgcn_wmma_f32_16x16x4_f32(false, a, false, bt0, (short)0, acc0, false, false);
        acc1 = __builtin_amdgcn_wmma_f32_16x16x4_f32(false, a, false, bt1, (short)0, acc1, false, false);
        acc2 = __builtin_amdgcn_wmma_f32_16x16x4_f32(false, a, false, bt2, (short)0, acc2, false, false);
    }

    // ---- bias + relu + store in (b, c, s) layout ----
    const int mh = (lane >= 16) ? 8 : 0;
#pragma unroll
    for (int r = 0; r < 8; ++r) {
        const int mm = r + mh;
        const int pp = tile * 16 + mm;
        const int bb = pp / S_DIM;
        const int ss = pp % S_DIM;
#pragma unroll
        for (int t = 0; t < 3; ++t) {
            const int cch = t * 16 + n0;
            if (cch < C_OUT) {
                float v = (t == 0 ? acc0[r] : (t == 1 ? acc1[r] : acc2[r])) + b2s[cch];
                Y[((size_t)bb * C_OUT + cch) * S_DIM + ss] = fmaxf(v, 0.0f);
            }
        }
    }
}

// ---------------------------------------------------------------------------
// Kernel 3: per-(b,c) spatial argmax (first-max tie-break), set mask column
// ---------------------------------------------------------------------------
__global__ void __launch_bounds__(128) argmax_kernel(const float* __restrict__ Y,
                                                     float* __restrict__ colmask) {
    const int row = blockIdx.x;                  // 0..2751
    const float* yr = Y + (size_t)row * S_DIM;
    float best = -INFINITY; int bi = 0;
    for (int s = threadIdx.x; s < S_DIM; s += 128) {
        float v = yr[s];
        if (v > best) { best = v; bi = s; }      // ascending scan -> first max kept
    }
    __shared__ float sv[128];
    __shared__ int   si[128];
    sv[threadIdx.x] = best; si[threadIdx.x] = bi;
    __syncthreads();
    for (int o = 64; o > 0; o >>= 1) {
        if (threadIdx.x < o) {
            float v2 = sv[threadIdx.x + o]; int i2 = si[threadIdx.x + o];
            if (v2 > sv[threadIdx.x] ||
                (v2 == sv[threadIdx.x] && i2 < si[threadIdx.x])) {
                sv[threadIdx.x] = v2; si[threadIdx.x] = i2;
            }
        }
        __syncthreads();
    }
    if (threadIdx.x == 0) colmask[si[0]] = 1.0f; // union of argmaxes (benign race)
}

// ---------------------------------------------------------------------------
// Kernel 4: separable 17-tap Gaussian blur (sigma=2), reflect ('symmetric')
// single block; pass H then W with an in-block sync
// ---------------------------------------------------------------------------
__device__ __forceinline__ int refl(int t, int n) {
    if (t < 0) t = -t - 1;
    if (t >= n) t = 2 * n - 1 - t;   // radius 8 < 50, one fold suffices
    return t;
}

__global__ void __launch_bounds__(256) blur_kernel(const float* __restrict__ mask,
                                                   float* __restrict__ tmp,
                                                   float* __restrict__ g) {
    float k[17];
    float ksum = 0.0f;
    for (int i = 0; i < 17; ++i) {
        float t = (float)(i - 8) * 0.5f;         // x / sigma, sigma = 2
        k[i] = expf(-0.5f * t * t);
        ksum += k[i];
    }
    for (int i = 0; i < 17; ++i) k[i] /= ksum;

    for (int i = threadIdx.x; i < S_DIM; i += 256) {   // blur along H (rows)
        int r = i / 50, c = i % 50;
        float acc = 0.0f;
        for (int j = 0; j < 17; ++j) acc += k[j] * mask[refl(r + j - 8, 50) * 50 + c];
        tmp[i] = acc;
    }
    __syncthreads();
    for (int i = threadIdx.x; i < S_DIM; i += 256) {   // blur along W (cols)
        int r = i / 50, c = i % 50;
        float acc = 0.0f;
        for (int j = 0; j < 17; ++j) acc += k[j] * tmp[r * 50 + refl(c + j - 8, 50)];
        g[i] = acc;
    }
}

// ---------------------------------------------------------------------------
// Kernel 5: out[b,c] = mean_s( Y[b,c,s] * g[s] )
// ---------------------------------------------------------------------------
__global__ void __launch_bounds__(256) reduce_kernel(const float* __restrict__ Y,
                                                     const float* __restrict__ g,
                                                     float* __restrict__ out) {
    const int row = blockIdx.x;                  // 0..2751
    const float* yr = Y + (size_t)row * S_DIM;
    float acc = 0.0f;
    for (int s = threadIdx.x; s < S_DIM; s += 256) acc = fmaf(yr[s], g[s], acc);
    __shared__ float sd[256];
    sd[threadIdx.x] = acc;
    __syncthreads();
    for (int o = 128; o > 0; o >>= 1) {
        if (threadIdx.x < o) sd[threadIdx.x] += sd[threadIdx.x + o];
        __syncthreads();
    }
    if (threadIdx.x == 0) out[row] = sd[0] * (1.0f / (float)S_DIM);
}

// ---------------------------------------------------------------------------
extern "C" void kernel_launch(void* const* d_in, const int* in_sizes, int n_in,
                              void* d_out, int out_size, void* d_ws, size_t ws_size,
                              hipStream_t stream) {
    const float* x  = (const float*)d_in[0];
    const float* w1 = (const float*)d_in[1];
    const float* b1 = (const float*)d_in[2];
    const float* w2 = (const float*)d_in[3];
    const float* b2 = (const float*)d_in[4];
    float* out = (float*)d_out;

    char* ws = (char*)d_ws;
    const size_t Y_BYTES = (size_t)B_DIM * C_OUT * S_DIM * sizeof(float); // 27,520,000
    float* Y       = (float*)(ws);
    float* colmask = (float*)(ws + Y_BYTES);
    float* tmp     = (float*)(ws + Y_BYTES + S_DIM * sizeof(float));
    float* g       = (float*)(ws + Y_BYTES + 2 * S_DIM * sizeof(float));

    init_mask_kernel<<<(S_DIM + 255) / 256, 256, 0, stream>>>(colmask);
    mlp_wmma_kernel<<<N_TILES / 8, 256, 0, stream>>>(x, w1, b1, w2, b2, Y);
    argmax_kernel<<<B_DIM * C_OUT, 128, 0, stream>>>(Y, colmask);
    blur_kernel<<<1, 256, 0, stream>>>(colmask, tmp, g);
    reduce_kernel<<<B_DIM * C_OUT, 256, 0, stream>>>(Y, g, out);
}